// Hard_Extract_Weight_Sum_64836826301210
// MI455X (gfx1250) — compile-verified
//
#include <hip/hip_runtime.h>
#include <hip/hip_bf16.h>

typedef __attribute__((ext_vector_type(2))) float v2f;
typedef __attribute__((ext_vector_type(8))) float v8f;

#define SEQ   512
#define DIM   768
#define HEADS 12

// ---------------------------------------------------------------------------
// Phase 0: zero the attended_by scratch (harness does not re-zero d_ws).
// ---------------------------------------------------------------------------
__global__ void zero_kernel(float* __restrict__ p, int n) {
  int i = blockIdx.x * blockDim.x + threadIdx.x;
  if (i < n) p[i] = 0.0f;
}

// ---------------------------------------------------------------------------
// Phase 1: column sums of each (b,h) 512x512 slice via V_WMMA_F32_16X16X4_F32
// with A = ones (so D[m,n] = sum_k B[k,n]); subtract diagonal, scale by 1/12,
// atomically accumulate into ab[b, col].  Pure HBM-streaming kernel: 805 MB
// read -> ~35 us floor at 23.3 TB/s.  One block per (b,h); 8 waves * 64 cols.
// ---------------------------------------------------------------------------
__global__ __launch_bounds__(256)
void colsum_wmma_kernel(const float* __restrict__ atten, float* __restrict__ ab) {
  const int bh = blockIdx.x;
  const int b  = bh / HEADS;
  const float* __restrict__ M = atten + (size_t)bh * SEQ * SEQ;

  const int lane = threadIdx.x & 31;
  const int wave = threadIdx.x >> 5;          // 0..7, each owns 64 columns
  const int nsub = lane & 15;                 // N within 16-col tile
  const int krow = (lane >> 4) << 1;          // row offset 0 or 2 within K=4 slab

  v2f ones; ones[0] = 1.0f; ones[1] = 1.0f;   // A matrix: all ones

  for (int g = 0; g < 4; ++g) {
    const int col = wave * 64 + g * 16 + nsub;
    v8f acc = {};
    const float* p = M + (size_t)krow * SEQ + col;
    for (int k0 = 0; k0 < SEQ; k0 += 4) {
      v2f bm;
      bm[0] = p[0];                           // row k0+krow,   this column
      bm[1] = p[SEQ];                         // row k0+krow+1, this column
      __builtin_prefetch(p + 16 * SEQ, 0, 1); // stream ahead (global_prefetch_b8)
      // D = A(ones) x B + C  ==> every row of D = running column sums
      acc = __builtin_amdgcn_wmma_f32_16x16x4_f32(
          /*neg_a=*/false, ones, /*neg_b=*/false, bm,
          /*c_mod=*/(short)0, acc, /*reuse_a=*/false, /*reuse_b=*/false);
      p += 4 * SEQ;
    }
    // C/D layout: VGPR0, lanes 0..15 hold D[M=0, N=lane] = colsum(col)
    if (lane < 16) {
      const float diag = M[(size_t)col * SEQ + col];
      atomicAdd(&ab[(size_t)b * SEQ + col], (acc[0] - diag) * (1.0f / 12.0f));
    }
  }
}

// ---------------------------------------------------------------------------
// Phase 2: per-batch selection.  Top (index-2) of ab[1:] are kept (plus cls),
// the complementary (SEQ-index+1) tokens get softmax weights; one streaming
// pass over x[b] copies kept rows to their sorted slots and accumulates the
// weighted-mean token.  One block (256 threads) per batch.
// ---------------------------------------------------------------------------
__global__ __launch_bounds__(256)
void select_kernel(const float* __restrict__ x, const float* __restrict__ abg,
                   const int* __restrict__ index_p, float* __restrict__ out) {
  const int b   = blockIdx.x;
  const int tid = threadIdx.x;
  const int index  = *index_p;            // 256
  const int keepN  = index - 2;           // 254 top tokens (excluding cls)
  const int kOther = SEQ - index + 1;     // 257 complementary tokens

  __shared__ float ab[SEQ];
  __shared__ float w[SEQ];
  __shared__ int   flag[SEQ];
  __shared__ int   pos[SEQ];
  __shared__ float red[2];

  for (int j = tid; j < SEQ; j += 256) ab[j] = abg[(size_t)b * SEQ + j];
  __syncthreads();

  // Rank each token j>=1 among ab[1:]; stable tie-break (lower index wins).
  for (int j = tid; j < SEQ; j += 256) {
    if (j == 0) { flag[0] = 0; continue; }
    const float v = ab[j];
    int rank = 0;
    for (int i = 1; i < SEQ; ++i) {
      const float u = ab[i];
      rank += (u > v || (u == v && i < j)) ? 1 : 0;
    }
    flag[j] = (rank < keepN) ? 1 : 0;
  }
  __syncthreads();

  // Output slot for kept tokens = 1 + (#kept tokens with smaller index).
  for (int j = tid; j < SEQ; j += 256) {
    if (flag[j]) {
      int p = 1;                          // slot 0 is the cls token
      for (int i = 1; i < j; ++i) p += flag[i];
      pos[j] = p;
    } else {
      pos[j] = -1;
    }
  }
  // Softmax stats over the "other" set (511 iterations: negligible).
  if (tid == 0) {
    float mx = -3.4e38f;
    for (int i = 1; i < SEQ; ++i) if (!flag[i]) mx = fmaxf(mx, ab[i]);
    float se = 0.0f;
    for (int i = 1; i < SEQ; ++i) if (!flag[i]) se += __expf(ab[i] - mx);
    red[0] = mx; red[1] = se;
  }
  __syncthreads();

  const float wscale = 1.0f / ((float)kOther * red[1]);   // softmax / mean
  for (int j = tid; j < SEQ; j += 256)
    w[j] = (j >= 1 && !flag[j]) ? __expf(ab[j] - red[0]) * wscale : 0.0f;
  __syncthreads();

  const float* __restrict__ xb = x + (size_t)b * SEQ * DIM;
  float* __restrict__ ob = out + (size_t)b * index * DIM;

  for (int d = tid; d < DIM; d += 256) ob[d] = xb[d];     // cls row

  // Each thread owns 3 columns (256*3 = 768); single pass over all rows.
  float a0 = 0.f, a1 = 0.f, a2 = 0.f;
  const int d0 = tid, d1 = tid + 256, d2 = tid + 512;
  for (int j = 1; j < SEQ; ++j) {
    const float v0 = xb[(size_t)j * DIM + d0];
    const float v1 = xb[(size_t)j * DIM + d1];
    const float v2 = xb[(size_t)j * DIM + d2];
    const int p = pos[j];
    if (p >= 0) {
      ob[(size_t)p * DIM + d0] = v0;
      ob[(size_t)p * DIM + d1] = v1;
      ob[(size_t)p * DIM + d2] = v2;
    } else {
      const float ww = w[j];
      a0 += ww * v0; a1 += ww * v1; a2 += ww * v2;
    }
  }
  ob[(size_t)(index - 1) * DIM + d0] = a0;
  ob[(size_t)(index - 1) * DIM + d1] = a1;
  ob[(size_t)(index - 1) * DIM + d2] = a2;
}

// ---------------------------------------------------------------------------
extern "C" void kernel_launch(void* const* d_in, const int* in_sizes, int n_in,
                              void* d_out, int out_size, void* d_ws, size_t ws_size,
                              hipStream_t stream) {
  const float* x       = (const float*)d_in[0];
  const float* atten   = (const float*)d_in[1];
  const int*   index_p = (const int*)d_in[2];
  float*       out     = (float*)d_out;
  float*       ab      = (float*)d_ws;                 // B*SEQ f32 scratch

  const int B  = in_sizes[0] / (SEQ * DIM);            // 64
  const int BH = in_sizes[1] / (SEQ * SEQ);            // 768

  const int nz = B * SEQ;
  zero_kernel<<<(nz + 255) / 256, 256, 0, stream>>>(ab, nz);
  colsum_wmma_kernel<<<BH, 256, 0, stream>>>(atten, ab);
  select_kernel<<<B, 256, 0, stream>>>(x, ab, index_p, out);
}